// Net_58961311040074
// MI455X (gfx1250) — compile-verified
//
#include <hip/hip_runtime.h>
#include <math.h>

typedef __attribute__((ext_vector_type(2))) float v2f;
typedef __attribute__((ext_vector_type(8))) float v8f;

#define WN_C       8
#define WN_SK      256
#define WN_NB      32
#define WN_TFULL   2048
#define WN_TEFF    1024     // receptive field of last timestep is 1021 -> 1024 window
#define WN_GUARD   128      // max dilation; zero-filled so h[t-d] reads return 0
#define WN_HSTRIDE 1168     // GUARD + TEFF + 16 pad (stride%64==16 -> bank-conflict-free)

// -------- CDNA5 WMMA: D(16x16 f32) = A(16x4 f32) * B(4x16 f32) + C ----------
static __device__ __forceinline__ v8f wmma_k4(v2f a, v2f b, v8f c) {
  return __builtin_amdgcn_wmma_f32_16x16x4_f32(false, a, false, b, (short)0, c,
                                               false, false);
}

// Branchless gate activations: v_exp_f32 + v_rcp_f32, no libm slow paths.
static __device__ __forceinline__ float fast_tanh(float x) {
  x = fminf(fmaxf(x, -15.f), 15.f);          // keep e^(2x) finite (no inf*0)
  float t = __expf(2.f * x);
  return (t - 1.f) * __builtin_amdgcn_rcpf(t + 1.f);
}
static __device__ __forceinline__ float fast_sigmoid(float x) {
  return __builtin_amdgcn_rcpf(1.f + __expf(-x));  // rcp(inf)=0, rcp(1)=1: safe
}

// Build A fragment for an 8x8 weight, K-half kh (cols kh*4..kh*4+3).
// ISA layout: lane l holds A[l%16][k0 + (l<16?0:2)] (.x) and +1 (.y).
// Rows 8..15 are zero-padded (guarded loads: never read OOB).
static __device__ __forceinline__ v2f afrag(const float* __restrict__ W,
                                            int ldr, int cst, int kh, int lane) {
  int o = lane & 15;
  int k = kh * 4 + ((lane >> 3) & 2);      // lanes<16 -> +0 ; lanes>=16 -> +2
  v2f a; a.x = 0.f; a.y = 0.f;
  if (o < 8) {
    a.x = W[o * ldr + k * cst];
    a.y = W[o * ldr + (k + 1) * cst];
  }
  return a;
}

// Build B fragment (4x16, K-half kh) from an LDS channel-major signal.
// hrow points at channel-0 row, element index already = GUARD + t0 (+shift).
static __device__ __forceinline__ v2f bfrag(const float* hrow, int kh,
                                            int lane, int tbase) {
  int c = kh * 4 + ((lane >> 3) & 2);
  int t = tbase + (lane & 15);
  v2f b;
  b.x = hrow[c * WN_HSTRIDE + t];
  b.y = hrow[(c + 1) * WN_HSTRIDE + t];
  return b;
}

// Re-layout a D fragment (rows 0..7 valid, lanes 0..15) into two B fragments
// for a chained matmul. Lanes>=16 need rows 2/3/6/7 from lane (l-16).
static __device__ __forceinline__ void d2b(v8f d, int lane, v2f& b0, v2f& b1) {
  int ls = lane & 15;
  float s2 = __shfl(d[2], ls, 32);
  float s3 = __shfl(d[3], ls, 32);
  float s6 = __shfl(d[6], ls, 32);
  float s7 = __shfl(d[7], ls, 32);
  bool lo = lane < 16;
  b0.x = lo ? d[0] : s2;
  b0.y = lo ? d[1] : s3;
  b1.x = lo ? d[4] : s6;
  b1.y = lo ? d[5] : s7;
}

// One workgroup (256 threads = 8 waves) per batch element; whole 32-block
// chain runs in-LDS with ping-pong buffers; one kernel launch total.
__global__ void __launch_bounds__(256)
wavenet_fused(const float* __restrict__ x,
              const float* __restrict__ w_in,  const float* __restrict__ b_in,
              const float* __restrict__ w_dil, const float* __restrict__ b_dil,
              const float* __restrict__ w_skip,const float* __restrict__ b_skip,
              const float* __restrict__ w_f,   const float* __restrict__ b_f,
              const float* __restrict__ w_g,   const float* __restrict__ b_g,
              const float* __restrict__ w_out, const float* __restrict__ b_out,
              const float* __restrict__ w_o1,  const float* __restrict__ b_o1,
              const float* __restrict__ w_o2,  const float* __restrict__ b_o2,
              float* __restrict__ out) {
  extern __shared__ float lds[];
  float* hA    = lds;                          // 8 * HSTRIDE
  float* hB    = lds + 8 * WN_HSTRIDE;         // 8 * HSTRIDE
  float* aLast = lds + 16 * WN_HSTRIDE;        // NB * 8
  float* svec  = aLast + WN_NB * WN_C;         // SK

  const int tid  = threadIdx.x;
  const int lane = tid & 31;
  const int wave = tid >> 5;
  const int bat  = blockIdx.x;

  // ---- init: zero guards of both buffers, input 1x1 conv over the window ----
  for (int i = tid; i < WN_C * WN_GUARD; i += 256) {
    int c = i / WN_GUARD, t = i % WN_GUARD;
    hA[c * WN_HSTRIDE + t] = 0.f;
    hB[c * WN_HSTRIDE + t] = 0.f;
  }
  for (int tw = tid; tw < WN_TEFF; tw += 256) {
    float xv = x[bat * WN_TFULL + (WN_TFULL - WN_TEFF) + tw];
#pragma unroll
    for (int c = 0; c < WN_C; ++c)
      hA[c * WN_HSTRIDE + WN_GUARD + tw] = w_in[c] * xv + b_in[c];
  }
  __syncthreads();

  float* hcur = hA;
  float* hnxt = hB;

  for (int i = 0; i < WN_NB; ++i) {
    const int d = 1 << (i & 7);
    // A fragments for this block's weights (uniform per wave)
    const float* Wd = w_dil + i * WN_C * WN_C * 2;   // (o, c, k2) row-major
    v2f aD0 = afrag(Wd + 0, 16, 2, 0, lane);  // W[...,0], c 0..3
    v2f aD1 = afrag(Wd + 0, 16, 2, 1, lane);  // W[...,0], c 4..7
    v2f aD2 = afrag(Wd + 1, 16, 2, 0, lane);  // W[...,1], c 0..3
    v2f aD3 = afrag(Wd + 1, 16, 2, 1, lane);  // W[...,1], c 4..7
    v2f aF0 = afrag(w_f  + i * 64, 8, 1, 0, lane);
    v2f aF1 = afrag(w_f  + i * 64, 8, 1, 1, lane);
    v2f aG0 = afrag(w_g  + i * 64, 8, 1, 0, lane);
    v2f aG1 = afrag(w_g  + i * 64, 8, 1, 1, lane);
    v2f aO0 = afrag(w_out + i * 64, 8, 1, 0, lane);
    v2f aO1 = afrag(w_out + i * 64, 8, 1, 1, lane);
    float bd[8], bfv[8], bgv[8], bov[8];
#pragma unroll
    for (int v = 0; v < 8; ++v) {
      bd[v]  = b_dil[i * 8 + v];
      bfv[v] = b_f  [i * 8 + v];
      bgv[v] = b_g  [i * 8 + v];
      bov[v] = b_out[i * 8 + v];
    }

    // 8 tiles of 16 timesteps per wave -> 1024 timesteps / 8 waves
#pragma unroll 2
    for (int tt = 0; tt < 8; ++tt) {
      const int t0 = wave * 128 + tt * 16;
      const int tg = WN_GUARD + t0;

      // dilated causal conv (K=16 as 4 chained K=4 WMMAs)
      v8f acc = {};
      acc = wmma_k4(aD0, bfrag(hcur, 0, lane, tg - d), acc);
      acc = wmma_k4(aD1, bfrag(hcur, 1, lane, tg - d), acc);
      acc = wmma_k4(aD2, bfrag(hcur, 0, lane, tg), acc);
      acc = wmma_k4(aD3, bfrag(hcur, 1, lane, tg), acc);
#pragma unroll
      for (int v = 0; v < 8; ++v) acc[v] = fmaxf(acc[v] + bd[v], 0.f);

      // stash a[:, T-1] for the skip path (one lane, once per block)
      if (lane < 16 && (t0 + lane) == (WN_TEFF - 1)) {
#pragma unroll
        for (int v = 0; v < 8; ++v) aLast[i * 8 + v] = acc[v];
      }

      // gates: f = tanh(Wf a + bf), g = sigmoid(Wg a + bg)
      v2f ab0, ab1;
      d2b(acc, lane, ab0, ab1);
      v8f fa = {}, ga = {};
      fa = wmma_k4(aF0, ab0, fa);
      fa = wmma_k4(aF1, ab1, fa);
      ga = wmma_k4(aG0, ab0, ga);
      ga = wmma_k4(aG1, ab1, ga);
      v8f fg;
#pragma unroll
      for (int v = 0; v < 8; ++v) {
        float fv = fast_tanh(fa[v] + bfv[v]);
        float gv = fast_sigmoid(ga[v] + bgv[v]);
        fg[v] = fv * gv;
      }

      // residual: h' = Wo (f*g) + bo + h
      v2f ob0, ob1;
      d2b(fg, lane, ob0, ob1);
      v8f oa = {};
      oa = wmma_k4(aO0, ob0, oa);
      oa = wmma_k4(aO1, ob1, oa);
      if (lane < 16) {
        const int t = tg + lane;
#pragma unroll
        for (int v = 0; v < 8; ++v)
          hnxt[v * WN_HSTRIDE + t] = oa[v] + bov[v] + hcur[v * WN_HSTRIDE + t];
      }
    }
    __syncthreads();
    float* tmp = hcur; hcur = hnxt; hnxt = tmp;
  }

  // ---- skip head: each thread owns one of 256 skip channels ----
  float sacc = 0.f;
#pragma unroll 4
  for (int i = 0; i < WN_NB; ++i) {
    const float4* wr = (const float4*)(w_skip + (i * WN_SK + tid) * WN_C);
    float4 w0 = wr[0], w1 = wr[1];
    const float* al = aLast + i * 8;
    sacc += w0.x * al[0] + w0.y * al[1] + w0.z * al[2] + w0.w * al[3]
          + w1.x * al[4] + w1.y * al[5] + w1.z * al[6] + w1.w * al[7]
          + b_skip[i * WN_SK + tid];
  }
  svec[tid] = fmaxf(sacc, 0.f);
  __syncthreads();

  float acc1 = b_o1[tid];
  const float4* wo1 = (const float4*)(w_o1 + tid * WN_SK);
#pragma unroll 8
  for (int j = 0; j < WN_SK; j += 4) {
    float4 w = wo1[j >> 2];
    acc1 += w.x * svec[j] + w.y * svec[j + 1] + w.z * svec[j + 2] + w.w * svec[j + 3];
  }
  acc1 = fmaxf(acc1, 0.f);
  __syncthreads();
  svec[tid] = acc1;
  __syncthreads();

  float acc2 = b_o2[tid];
  const float4* wo2 = (const float4*)(w_o2 + tid * WN_SK);
#pragma unroll 8
  for (int j = 0; j < WN_SK; j += 4) {
    float4 w = wo2[j >> 2];
    acc2 += w.x * svec[j] + w.y * svec[j + 1] + w.z * svec[j + 2] + w.w * svec[j + 3];
  }
  out[bat * WN_SK + tid] = acc2;
}

extern "C" void kernel_launch(void* const* d_in, const int* in_sizes, int n_in,
                              void* d_out, int out_size, void* d_ws, size_t ws_size,
                              hipStream_t stream) {
  (void)n_in; (void)d_ws; (void)ws_size; (void)out_size;
  const float* x      = (const float*)d_in[0];
  const float* w_in   = (const float*)d_in[1];
  const float* b_in   = (const float*)d_in[2];
  const float* w_dil  = (const float*)d_in[3];
  const float* b_dil  = (const float*)d_in[4];
  const float* w_skip = (const float*)d_in[5];
  const float* b_skip = (const float*)d_in[6];
  const float* w_f    = (const float*)d_in[7];
  const float* b_f    = (const float*)d_in[8];
  const float* w_g    = (const float*)d_in[9];
  const float* b_g    = (const float*)d_in[10];
  const float* w_out  = (const float*)d_in[11];
  const float* b_out  = (const float*)d_in[12];
  const float* w_o1   = (const float*)d_in[13];
  const float* b_o1   = (const float*)d_in[14];
  const float* w_o2   = (const float*)d_in[15];
  const float* b_o2   = (const float*)d_in[16];

  const int B = in_sizes[0] / WN_TFULL;   // 32
  const size_t shmem =
      (size_t)(16 * WN_HSTRIDE + WN_NB * WN_C + WN_SK) * sizeof(float); // ~75 KB

  wavenet_fused<<<B, 256, shmem, stream>>>(
      x, w_in, b_in, w_dil, b_dil, w_skip, b_skip,
      w_f, b_f, w_g, b_g, w_out, b_out,
      w_o1, b_o1, w_o2, b_o2, (float*)d_out);
}